// MLA_22436909154535
// MI455X (gfx1250) — compile-verified
//
#include <hip/hip_runtime.h>
#include <hip/hip_bf16.h>
#include <math.h>

// ---------------- problem constants ----------------
#define DIMX   2048
#define HCNT   16
#define NOPE   128
#define ROPE   64
#define VD     128
#define QR     1536
#define KVR    512
#define BCNT   4
#define L_TOT  4096
#define HDF    576          // KVR + ROPE
#define HDF4   144          // HDF / 4 (float4 units)
#define HDP    580          // padded LDS row stride (580 % 64 == 4 -> conflict free)
#define CHUNK  64           // KV positions per workgroup
#define NCH    (L_TOT / CHUNK)   // 64
#define SP     68           // S-tile LDS stride
#define SCALE  0.07216878364870323f   // 1/sqrt(192)
#define LOG1E4_OVER_32 0.28782313662425574f

typedef __attribute__((ext_vector_type(2))) float v2f;
typedef __attribute__((ext_vector_type(8))) float v8f;

__device__ __forceinline__ v8f wmma_f32_4(v2f a, v2f b, v8f c) {
  // D = A(16x4) * B(4x16) + C, fp32, wave32
  return __builtin_amdgcn_wmma_f32_16x16x4_f32(false, a, false, b, (short)0, c,
                                               false, false);
}

// ---- CDNA5 async global->LDS mover (ASYNCcnt-tracked, no VGPR round trip) --
__device__ __forceinline__ void async_copy_b128(void* lds_dst,
                                                const void* g_src) {
  // Low 32 bits of a generic pointer to __shared__ are the LDS byte offset
  // (flat->LDS mapping truncates to addr[31:0], ISA 10.2).
  uint32_t lds_off = (uint32_t)(uintptr_t)lds_dst;
  asm volatile("global_load_async_to_lds_b128 %0, %1, off"
               :
               : "v"(lds_off), "v"(g_src)
               : "memory");
}
__device__ __forceinline__ void wait_async_zero() {
  asm volatile("s_wait_asynccnt 0" ::: "memory");
}

// ---------------- kv cache copy: (B,4095,576) -> (B,4096,576) rows [0,4095) --
__global__ __launch_bounds__(256) void copy_cache_kernel(
    const float4* __restrict__ src, float4* __restrict__ dst) {
  const size_t PER_B_SRC = (size_t)4095 * HDF / 4;   // 589,680
  const size_t PER_B_DST = (size_t)L_TOT * HDF / 4;  // 589,824
  const size_t total = (size_t)BCNT * PER_B_SRC;
  for (size_t i = (size_t)blockIdx.x * blockDim.x + threadIdx.x; i < total;
       i += (size_t)gridDim.x * blockDim.x) {
    size_t b = i / PER_B_SRC, r = i - b * PER_B_SRC;
    dst[b * PER_B_DST + r] = src[i];
  }
}

// ---------------- split-K GEMV: out[b,j] = sum_i X[b,i] * W[i,j] ------------
// grid = (ceil(N/256), SK).  Writes partials; deterministic reduce pass below.
__global__ __launch_bounds__(256) void gemv_splitk_kernel(
    const float* __restrict__ X, const float* __restrict__ W,
    float* __restrict__ part, int K, int N) {
  __shared__ float xs[4 * 256];   // SK >= 8 for K<=2048 -> kb<=256
  const int sk = blockIdx.y;
  const int kb = K / gridDim.y;
  const int k0 = sk * kb;
  for (int i = threadIdx.x; i < BCNT * kb; i += 256) {
    int b = i / kb, ii = i - b * kb;
    xs[i] = X[(size_t)b * K + k0 + ii];
  }
  __syncthreads();
  const int j = blockIdx.x * 256 + threadIdx.x;
  if (j >= N) return;
  float a0 = 0.f, a1 = 0.f, a2 = 0.f, a3 = 0.f;
#pragma unroll 4
  for (int i = 0; i < kb; ++i) {
    float wv = W[(size_t)(k0 + i) * N + j];
    a0 += xs[i] * wv;
    a1 += xs[kb + i] * wv;
    a2 += xs[2 * kb + i] * wv;
    a3 += xs[3 * kb + i] * wv;
  }
  part[((size_t)sk * 4 + 0) * N + j] = a0;
  part[((size_t)sk * 4 + 1) * N + j] = a1;
  part[((size_t)sk * 4 + 2) * N + j] = a2;
  part[((size_t)sk * 4 + 3) * N + j] = a3;
}

__global__ __launch_bounds__(256) void gemv_reduce_kernel(
    const float* __restrict__ part, float* __restrict__ out, int N, int SK,
    size_t ob_stride) {
  const int j = blockIdx.x * 256 + threadIdx.x;
  if (j >= N) return;
  for (int b = 0; b < BCNT; ++b) {
    float acc = 0.f;
    for (int sk = 0; sk < SK; ++sk) acc += part[((size_t)sk * 4 + b) * N + j];
    out[(size_t)b * ob_stride + j] = acc;
  }
}

// ---------------- absorbed query build --------------------------------------
// q_full[b,h, 0:512]  = sum_d qnr[b, h*192 + d] * Wkv_up[r, h*256 + d]
// q_full[b,h,512:576] = rope(qnr[b, h*192 + 128 : +64], pos = 4095)
__global__ __launch_bounds__(256) void build_q_full_kernel(
    const float* __restrict__ qnr, const float* __restrict__ Wkv_up,
    float* __restrict__ qf) {
  __shared__ float qn[NOPE + ROPE];
  const int b = blockIdx.x >> 4;
  const int h = blockIdx.x & 15;
  const int tid = threadIdx.x;
  if (tid < NOPE + ROPE) qn[tid] = qnr[(size_t)b * 3072 + h * 192 + tid];
  __syncthreads();
  float* dst = qf + ((size_t)b * HCNT + h) * HDF;
  for (int r = tid; r < KVR; r += 256) {
    const float* wrow = Wkv_up + (size_t)r * 4096 + h * 256;
    float acc = 0.f;
#pragma unroll 4
    for (int d = 0; d < NOPE; ++d) acc += qn[d] * wrow[d];
    dst[r] = acc;
  }
  if (tid < 32) {
    float theta = __expf(-(float)tid * LOG1E4_OVER_32);
    float ang = 4095.0f * theta;
    float sn = sinf(ang), cn = cosf(ang);
    float x1 = qn[NOPE + tid], x2 = qn[NOPE + 32 + tid];
    dst[KVR + tid] = x1 * cn - x2 * sn;
    dst[KVR + 32 + tid] = x2 * cn + x1 * sn;
  }
}

// ---------------- flash-decode chunk kernel (WMMA core) ---------------------
// grid = (B, NCH); 256 threads = 8 wave32.  One WG per WGP (190KB LDS), so the
// tile fill uses the CDNA5 async global->LDS movers to hide fill latency.
__global__ __launch_bounds__(256) void attn_chunk_kernel(
    const float* __restrict__ kv,   // (B, 4096, 576) = d_out KV region
    const float* __restrict__ qf,   // (B, 16, 576)
    float* __restrict__ tbuf,       // (B, NCH, 16, 512)
    float* __restrict__ mbuf,       // (B, NCH, 16)
    float* __restrict__ sbuf) {     // (B, NCH, 16)
  __shared__ float qs[HCNT * HDP];   //  37,120 B
  __shared__ float ks[CHUNK * HDP];  // 148,480 B
  __shared__ float Ss[HCNT * SP];    //   4,352 B
  __shared__ float ms[HCNT];

  const int b = blockIdx.x;
  const int ch = blockIdx.y;
  const int tid = threadIdx.x;
  const int lane = tid & 31;
  const int w = tid >> 5;

  // ---- async-fill Q (16x576) and K chunk (64x576) into padded LDS ----
  {
    const float4* qsrc4 = (const float4*)(qf + (size_t)b * HCNT * HDF);
    for (int i = tid; i < HCNT * HDF4; i += 256) {
      int row = i / HDF4, col = i - row * HDF4;
      async_copy_b128(&qs[row * HDP + col * 4], qsrc4 + i);
    }
    const float4* ksrc4 =
        (const float4*)(kv + ((size_t)b * L_TOT + (size_t)ch * CHUNK) * HDF);
    for (int i = tid; i < CHUNK * HDF4; i += 256) {
      int row = i / HDF4, col = i - row * HDF4;
      async_copy_b128(&ks[row * HDP + col * 4], ksrc4 + i);
    }
    wait_async_zero();   // drain this wave's ASYNCcnt before the barrier
  }
  __syncthreads();

  // ---- rope the last 64 features of each K row in-place ----
  for (int t = tid; t < CHUNK * 32; t += 256) {
    int row = t >> 5, i = t & 31;
    float theta = __expf(-(float)i * LOG1E4_OVER_32);
    float ang = (float)(ch * CHUNK + row) * theta;
    float sn = sinf(ang), cn = cosf(ang);
    float x1 = ks[row * HDP + KVR + i];
    float x2 = ks[row * HDP + KVR + 32 + i];
    ks[row * HDP + KVR + i] = x1 * cn - x2 * sn;
    ks[row * HDP + KVR + 32 + i] = x2 * cn + x1 * sn;
  }
  __syncthreads();

  const int mrow = lane & 15;            // A-matrix row / B-matrix column
  const int kk = (lane >> 4) << 1;       // K sub-offset per half-wave (0 or 2)
  const int crow0 = (lane >> 4) << 3;    // C/D layout: rows v / v+8
  const int ccol = lane & 15;

  // ---- S = Q(16x576) x K^T(576x64), K-dim split over 2 wave groups ----
  {
    const int tile = w & 3;    // which 16-column tile of S
    const int khalf = w >> 2;  // which 288-wide K half
    const int n0 = tile * 16;
    v8f c = {0.f, 0.f, 0.f, 0.f, 0.f, 0.f, 0.f, 0.f};
    const int kbeg = khalf * 288;
    for (int k = kbeg; k < kbeg + 288; k += 4) {
      v2f a, bb;
      a.x = qs[mrow * HDP + k + kk];
      a.y = qs[mrow * HDP + k + kk + 1];
      bb.x = ks[(n0 + mrow) * HDP + k + kk];
      bb.y = ks[(n0 + mrow) * HDP + k + kk + 1];
      c = wmma_f32_4(a, bb, c);
    }
    if (khalf == 0) {
#pragma unroll
      for (int v = 0; v < 8; ++v) Ss[(v + crow0) * SP + n0 + ccol] = c[v];
    }
    __syncthreads();
    if (khalf == 1) {
#pragma unroll
      for (int v = 0; v < 8; ++v) Ss[(v + crow0) * SP + n0 + ccol] += c[v];
    }
    __syncthreads();
  }

  // ---- chunk-local softmax stats, convert S -> P in place ----
  if (tid < HCNT) {
    float mx = -3.4e38f;
    for (int c2 = 0; c2 < CHUNK; ++c2) mx = fmaxf(mx, Ss[tid * SP + c2]);
    ms[tid] = mx;
  }
  __syncthreads();
  for (int t = tid; t < HCNT * CHUNK; t += 256) {
    int row = t / CHUNK, c2 = t - row * CHUNK;
    Ss[row * SP + c2] = __expf((Ss[row * SP + c2] - ms[row]) * SCALE);
  }
  __syncthreads();
  if (tid < HCNT) {
    float s = 0.f;
    for (int c2 = 0; c2 < CHUNK; ++c2) s += Ss[tid * SP + c2];
    mbuf[((size_t)b * NCH + ch) * HCNT + tid] = ms[tid] * SCALE;
    sbuf[((size_t)b * NCH + ch) * HCNT + tid] = s;
  }

  // ---- T = P(16x64) x KV_nope(64x512); 8 waves x 4 N-tiles ----
  {
    float* tdst = tbuf + ((size_t)b * NCH + ch) * HCNT * KVR;
#pragma unroll
    for (int tt = 0; tt < 4; ++tt) {
      const int n0 = (w * 4 + tt) * 16;
      v8f c = {0.f, 0.f, 0.f, 0.f, 0.f, 0.f, 0.f, 0.f};
      for (int k = 0; k < CHUNK; k += 4) {
        v2f a, bb;
        a.x = Ss[mrow * SP + k + kk];
        a.y = Ss[mrow * SP + k + kk + 1];
        bb.x = ks[(k + kk) * HDP + n0 + mrow];
        bb.y = ks[(k + kk + 1) * HDP + n0 + mrow];
        c = wmma_f32_4(a, bb, c);
      }
#pragma unroll
      for (int v = 0; v < 8; ++v)
        tdst[(size_t)(v + crow0) * KVR + n0 + ccol] = c[v];
    }
  }
}

// ---------------- combine chunks + absorbed V up-projection -----------------
// grid = B*H blocks; y[b, h*128 + c] = sum_r T_final[r] * Wkv_up[r, h*256+128+c]
__global__ __launch_bounds__(256) void combine_kernel(
    const float* __restrict__ tbuf, const float* __restrict__ mbuf,
    const float* __restrict__ sbuf, const float* __restrict__ Wkv_up,
    float* __restrict__ y) {
  __shared__ float wch[NCH];
  __shared__ float tf[KVR];
  __shared__ float dsh;
  const int b = blockIdx.x >> 4;
  const int h = blockIdx.x & 15;
  const int tid = threadIdx.x;
  if (tid == 0) {
    float mmax = -3.4e38f;
    for (int c = 0; c < NCH; ++c)
      mmax = fmaxf(mmax, mbuf[((size_t)b * NCH + c) * HCNT + h]);
    float denom = 0.f;
    for (int c = 0; c < NCH; ++c) {
      float wc = __expf(mbuf[((size_t)b * NCH + c) * HCNT + h] - mmax);
      wch[c] = wc;
      denom += wc * sbuf[((size_t)b * NCH + c) * HCNT + h];
    }
    dsh = denom;
  }
  __syncthreads();
  const float inv = 1.0f / dsh;
  for (int r = tid; r < KVR; r += 256) {
    float acc = 0.f;
    for (int c = 0; c < NCH; ++c)
      acc += wch[c] * tbuf[(((size_t)b * NCH + c) * HCNT + h) * KVR + r];
    tf[r] = acc * inv;
  }
  __syncthreads();
  if (tid < VD) {
    float acc = 0.f;
#pragma unroll 4
    for (int r = 0; r < KVR; ++r)
      acc += tf[r] * Wkv_up[(size_t)r * 4096 + h * 256 + NOPE + tid];
    y[(size_t)b * (HCNT * VD) + h * VD + tid] = acc;
  }
}

// ---------------- host-side orchestration -----------------------------------
extern "C" void kernel_launch(void* const* d_in, const int* in_sizes, int n_in,
                              void* d_out, int out_size, void* d_ws,
                              size_t ws_size, hipStream_t stream) {
  (void)in_sizes; (void)n_in; (void)out_size; (void)ws_size;
  const float* x        = (const float*)d_in[0];  // (4,1,2048)
  const float* kvprev   = (const float*)d_in[1];  // (4,4095,576)
  const float* Wq_down  = (const float*)d_in[2];  // (2048,1536)
  const float* Wkv_down = (const float*)d_in[3];  // (2048,576)
  const float* Wq_up    = (const float*)d_in[4];  // (1536,3072)
  const float* Wkv_up   = (const float*)d_in[5];  // (512,4096)
  const float* Wo       = (const float*)d_in[6];  // (2048,2048)

  float* out   = (float*)d_out;                   // (4,1,2048)
  float* kvout = out + (size_t)BCNT * DIMX;       // (4,4096,576)

  float* ws     = (float*)d_ws;
  float* q_lora = ws;               // 4*1536          =     6,144
  float* qnr    = ws + 6144;        // 4*3072          =    12,288
  float* qf     = ws + 18432;       // 4*16*576        =    36,864
  float* yv     = ws + 55296;       // 4*2048          =     8,192
  float* mb     = ws + 63488;       // 4*64*16         =     4,096
  float* sb     = ws + 67584;       // 4*64*16         =     4,096
  float* part   = ws + 71680;       // 8*4*3072 max    =    98,304
  float* tb     = ws + 169984;      // 4*64*16*512     = 2,097,152

  // 1) kv_lora cache: copy old rows, project new row (row 4095)
  copy_cache_kernel<<<4096, 256, 0, stream>>>((const float4*)kvprev,
                                              (float4*)kvout);
  gemv_splitk_kernel<<<dim3(3, 8), 256, 0, stream>>>(x, Wkv_down, part, DIMX,
                                                     HDF);
  gemv_reduce_kernel<<<3, 256, 0, stream>>>(part, kvout + (size_t)4095 * HDF,
                                            HDF, 8, (size_t)L_TOT * HDF);

  // 2) q_lora = x @ Wq_down ; qnr = q_lora @ Wq_up
  gemv_splitk_kernel<<<dim3(6, 8), 256, 0, stream>>>(x, Wq_down, part, DIMX, QR);
  gemv_reduce_kernel<<<6, 256, 0, stream>>>(part, q_lora, QR, 8, QR);
  gemv_splitk_kernel<<<dim3(12, 8), 256, 0, stream>>>(q_lora, Wq_up, part, QR,
                                                      3072);
  gemv_reduce_kernel<<<12, 256, 0, stream>>>(part, qnr, 3072, 8, 3072);

  // 3) absorbed q_full (q_nope @ Wk^T  ||  rope(q_rope))
  build_q_full_kernel<<<BCNT * HCNT, 256, 0, stream>>>(qnr, Wkv_up, qf);

  // 4) flash-decode over 64-position chunks (WMMA core, async LDS fill)
  attn_chunk_kernel<<<dim3(BCNT, NCH), 256, 0, stream>>>(kvout, qf, tb, mb, sb);

  // 5) combine + absorbed V up-projection -> y (4,2048)
  combine_kernel<<<BCNT * HCNT, 256, 0, stream>>>(tb, mb, sb, Wkv_up, yv);

  // 6) out = y @ Wo
  gemv_splitk_kernel<<<dim3(8, 8), 256, 0, stream>>>(yv, Wo, part, DIMX, DIMX);
  gemv_reduce_kernel<<<8, 256, 0, stream>>>(part, out, DIMX, 8, DIMX);
}